// ZINC20ContextOnly_89550068122082
// MI455X (gfx1250) — compile-verified
//
#include <hip/hip_runtime.h>
#include <math.h>

#define NN      20000
#define NE      320000
#define ETOT    (NE + NN)      // edges + self loops
#define CDIM    128            // out channels per head (all layers)
#define LRELU   0.2f
#define BN_EPS  1e-5f

typedef __attribute__((ext_vector_type(2))) float v2f;
typedef __attribute__((ext_vector_type(8))) float v8f;

// ---------------------------------------------------------------------------
// C[M,N] = A[M,K] @ B[K,N] (+bias)(+relu), fp32 WMMA 16x16x4.
// One wave computes a (16*MT) x (16*NT) tile, software-pipelined (fragments
// for step k+4 are loaded before the WMMAs of step k issue). K,N are template
// constants so every fragment address is a compile-time IOFFSET.
// REQUIRES: M%(16*MT)==0, N%(16*NT)==0, K%4==0, K>=8.
// A-fragment layout (ISA 7.12.2): lane<16 holds A[lid][k,k+1], lane>=16 holds
// A[lid][k+2,k+3]. B mirrors. C: VGPR j -> row j (+8 for upper half), col lid.
// ---------------------------------------------------------------------------
template <int K, int N, int MT, int NT>
__global__ void gemm_wmma(const float* __restrict__ A,
                          const float* __restrict__ B,
                          float* __restrict__ C,
                          const float* __restrict__ bias,
                          int relu)
{
    const int tileN = blockIdx.x * (16 * NT);
    const int tileM = blockIdx.y * (16 * MT);
    const int lane  = threadIdx.x;      // 0..31, full wave
    const int half  = lane >> 4;
    const int lid   = lane & 15;

    const float* Ap = A + (long)(tileM + lid) * K + 2 * half;
    const float* Bp = B + (long)(2 * half) * N + tileN + lid;

    v8f acc[MT][NT];
    #pragma unroll
    for (int m = 0; m < MT; ++m)
        #pragma unroll
        for (int t = 0; t < NT; ++t) acc[m][t] = (v8f){};

    // prologue: fragments for k = 0
    v2f a[MT], b[NT];
    #pragma unroll
    for (int m = 0; m < MT; ++m) { a[m][0] = Ap[m * 16 * K]; a[m][1] = Ap[m * 16 * K + 1]; }
    #pragma unroll
    for (int t = 0; t < NT; ++t) { b[t][0] = Bp[t * 16];     b[t][1] = Bp[t * 16 + N]; }

    #pragma unroll 2
    for (int k = 0; k < K - 4; k += 4) {
        Ap += 4;
        Bp += 4 * N;
        // prefetch next iteration's fragments (no dependency on current WMMAs)
        v2f an[MT], bn[NT];
        #pragma unroll
        for (int m = 0; m < MT; ++m) { an[m][0] = Ap[m * 16 * K]; an[m][1] = Ap[m * 16 * K + 1]; }
        #pragma unroll
        for (int t = 0; t < NT; ++t) { bn[t][0] = Bp[t * 16];     bn[t][1] = Bp[t * 16 + N]; }
        // consume current fragments
        #pragma unroll
        for (int m = 0; m < MT; ++m)
            #pragma unroll
            for (int t = 0; t < NT; ++t)
                acc[m][t] = __builtin_amdgcn_wmma_f32_16x16x4_f32(
                                false, a[m], false, b[t], (short)0, acc[m][t], false, false);
        #pragma unroll
        for (int m = 0; m < MT; ++m) a[m] = an[m];
        #pragma unroll
        for (int t = 0; t < NT; ++t) b[t] = bn[t];
    }
    // epilogue: last k-step
    #pragma unroll
    for (int m = 0; m < MT; ++m)
        #pragma unroll
        for (int t = 0; t < NT; ++t)
            acc[m][t] = __builtin_amdgcn_wmma_f32_16x16x4_f32(
                            false, a[m], false, b[t], (short)0, acc[m][t], false, false);

    #pragma unroll
    for (int m = 0; m < MT; ++m) {
        #pragma unroll
        for (int t = 0; t < NT; ++t) {
            const int col = tileN + t * 16 + lid;
            const float bb = bias ? bias[col] : 0.0f;
            #pragma unroll
            for (int j = 0; j < 8; ++j) {
                const int row = tileM + m * 16 + j + 8 * half;
                float v = acc[m][t][j] + bb;
                if (relu) v = fmaxf(v, 0.0f);
                C[(long)row * N + col] = v;
            }
        }
    }
}

// ---------------------------------------------------------------------------
// Zero-pad a [sr,sc] row-major matrix into [dr,dc].
// ---------------------------------------------------------------------------
__global__ void pad2d(const float* __restrict__ src, float* __restrict__ dst,
                      int sr, int sc, int dr, int dc)
{
    const long gid = (long)blockIdx.x * blockDim.x + threadIdx.x;
    if (gid >= (long)dr * dc) return;
    const int r = (int)(gid / dc);
    const int c = (int)(gid % dc);
    dst[gid] = (r < sr && c < sc) ? src[(long)r * sc + c] : 0.0f;
}

// slice first 9 of 16 columns into d_out
__global__ void slice9(const float* __restrict__ src, float* __restrict__ dst)
{
    const long gid = (long)blockIdx.x * blockDim.x + threadIdx.x;
    if (gid >= (long)NN * 9) return;
    const int r = (int)(gid / 9);
    const int c = (int)(gid % 9);
    dst[gid] = src[(long)r * 16 + c];
}

// ---------------------------------------------------------------------------
// s[n,h] = sum_c xl[n,h,c]*a_src[h,c] ; d likewise. One wave per (n,h).
// ---------------------------------------------------------------------------
__global__ void sd_kernel(const float* __restrict__ xl,
                          const float* __restrict__ a_src,
                          const float* __restrict__ a_dst,
                          float* __restrict__ s, float* __restrict__ d, int H)
{
    const int n    = blockIdx.x;
    const int h    = threadIdx.x >> 5;
    const int lane = threadIdx.x & 31;
    const float* xp = xl    + (long)n * H * CDIM + h * CDIM;
    const float* ap = a_src + h * CDIM;
    const float* bp = a_dst + h * CDIM;

    const int c = lane * 4;                      // 32 lanes * 4 = 128 = CDIM
    float4 xv = *(const float4*)(xp + c);
    float4 av = *(const float4*)(ap + c);
    float4 bv = *(const float4*)(bp + c);
    float ss = xv.x*av.x + xv.y*av.y + xv.z*av.z + xv.w*av.w;
    float dd = xv.x*bv.x + xv.y*bv.y + xv.z*bv.z + xv.w*bv.w;
    #pragma unroll
    for (int off = 16; off > 0; off >>= 1) {
        ss += __shfl_down(ss, off, 32);
        dd += __shfl_down(dd, off, 32);
    }
    if (lane == 0) { s[n * H + h] = ss; d[n * H + h] = dd; }
}

// ---------------------------------------------------------------------------
__global__ void init_kernel(float* __restrict__ out, const float* __restrict__ bias,
                            float* __restrict__ emax, float* __restrict__ denom,
                            int HC, int H)
{
    const long gid = (long)blockIdx.x * blockDim.x + threadIdx.x;
    if (gid < (long)NN * HC) out[gid] = bias[gid % HC];
    if (gid < (long)NN * H) { emax[gid] = -INFINITY; denom[gid] = 0.0f; }
}

__device__ __forceinline__ void edge_sd(const int* __restrict__ ei, int e,
                                        int& src, int& dst)
{
    if (e < NE) { src = ei[e]; dst = ei[NE + e]; }
    else        { src = dst = e - NE; }          // self loops appended
}

__device__ __forceinline__ float atomicMaxF(float* addr, float val)
{
    if (val >= 0.0f)
        return __int_as_float(atomicMax((int*)addr, __float_as_int(val)));
    return __uint_as_float(atomicMin((unsigned int*)addr, __float_as_uint(val)));
}

__global__ void edge_max(const int* __restrict__ ei,
                         const float* __restrict__ s, const float* __restrict__ d,
                         float* __restrict__ emax, int H)
{
    const int e = blockIdx.x * blockDim.x + threadIdx.x;
    if (e >= ETOT) return;
    int src, dst; edge_sd(ei, e, src, dst);
    for (int h = 0; h < H; ++h) {
        float v = s[src * H + h] + d[dst * H + h];
        v = (v > 0.0f) ? v : LRELU * v;
        atomicMaxF(&emax[dst * H + h], v);
    }
}

__global__ void edge_expsum(const int* __restrict__ ei,
                            const float* __restrict__ s, const float* __restrict__ d,
                            const float* __restrict__ emax, float* __restrict__ denom,
                            int H)
{
    const int e = blockIdx.x * blockDim.x + threadIdx.x;
    if (e >= ETOT) return;
    int src, dst; edge_sd(ei, e, src, dst);
    for (int h = 0; h < H; ++h) {
        float v = s[src * H + h] + d[dst * H + h];
        v = (v > 0.0f) ? v : LRELU * v;
        atomicAdd(&denom[dst * H + h], expf(v - emax[dst * H + h]));
    }
}

// out[dst, t] += alpha(e,h) * xl[src, t]  -- one thread per (edge, channel)
__global__ void edge_scatter(const int* __restrict__ ei,
                             const float* __restrict__ s, const float* __restrict__ d,
                             const float* __restrict__ emax, const float* __restrict__ denom,
                             const float* __restrict__ xl, float* __restrict__ out, int H)
{
    const long gid = (long)blockIdx.x * blockDim.x + threadIdx.x;
    const int  HC  = H * CDIM;
    if (gid >= (long)ETOT * HC) return;
    const int e = (int)(gid / HC);
    const int t = (int)(gid % HC);
    const int h = t >> 7;                        // t / 128
    int src, dst; edge_sd(ei, e, src, dst);
    float v = s[src * H + h] + d[dst * H + h];
    v = (v > 0.0f) ? v : LRELU * v;
    const float alpha = expf(v - emax[dst * H + h]) / (denom[dst * H + h] + 1e-16f);
    atomicAdd(&out[(long)dst * HC + t], alpha * xl[(long)src * HC + t]);
}

__global__ void bn_elu(const float* __restrict__ x,
                       const float* __restrict__ g, const float* __restrict__ be,
                       const float* __restrict__ m, const float* __restrict__ v,
                       float* __restrict__ y, int HC)
{
    const long gid = (long)blockIdx.x * blockDim.x + threadIdx.x;
    if (gid >= (long)NN * HC) return;
    const int j = (int)(gid % HC);
    const float scale = g[j] / sqrtf(v[j] + BN_EPS);
    const float t = (x[gid] - m[j]) * scale + be[j];
    y[gid] = (t > 0.0f) ? t : (expf(t) - 1.0f);
}

// ---------------------------------------------------------------------------
static void run_attention(const int* ei, const float* asr, const float* ads,
                          const float* bia, const float* gg, const float* bb,
                          const float* mm, const float* vv,
                          float* xl, float* aggr, float* bnout,
                          float* sbuf, float* dbuf, float* emax, float* denom,
                          int H, hipStream_t stream)
{
    const int HC = H * CDIM;
    sd_kernel<<<NN, H * 32, 0, stream>>>(xl, asr, ads, sbuf, dbuf, H);
    const long tot = (long)NN * HC;
    init_kernel<<<(int)((tot + 255) / 256), 256, 0, stream>>>(aggr, bia, emax, denom, HC, H);
    edge_max   <<<(ETOT + 255) / 256, 256, 0, stream>>>(ei, sbuf, dbuf, emax, H);
    edge_expsum<<<(ETOT + 255) / 256, 256, 0, stream>>>(ei, sbuf, dbuf, emax, denom, H);
    const long tot3 = (long)ETOT * HC;
    edge_scatter<<<(int)((tot3 + 255) / 256), 256, 0, stream>>>(
        ei, sbuf, dbuf, emax, denom, xl, aggr, H);
    bn_elu<<<(int)((tot + 255) / 256), 256, 0, stream>>>(aggr, gg, bb, mm, vv, bnout, HC);
}

extern "C" void kernel_launch(void* const* d_in, const int* in_sizes, int n_in,
                              void* d_out, int out_size, void* d_ws, size_t ws_size,
                              hipStream_t stream)
{
    const float* x  = (const float*)d_in[0];
    const int*   ei = (const int*)d_in[1];
    // layer param blocks start at 2, 10, 18 : W, a_src, a_dst, b, g, be, m, v
    const float *W[3], *asr[3], *ads[3], *bia[3], *gg[3], *bb[3], *mm[3], *vv[3];
    for (int L = 0; L < 3; ++L) {
        const int base = 2 + 8 * L;
        W[L]   = (const float*)d_in[base + 0];
        asr[L] = (const float*)d_in[base + 1];
        ads[L] = (const float*)d_in[base + 2];
        bia[L] = (const float*)d_in[base + 3];
        gg[L]  = (const float*)d_in[base + 4];
        bb[L]  = (const float*)d_in[base + 5];
        mm[L]  = (const float*)d_in[base + 6];
        vv[L]  = (const float*)d_in[base + 7];
    }
    const float* hw1 = (const float*)d_in[26];
    const float* hb1 = (const float*)d_in[27];
    const float* hw2 = (const float*)d_in[28];
    const float* hb2 = (const float*)d_in[29];

    float* ws = (float*)d_ws;
    const long BIG = (long)NN * 512;
    float* big0   = ws;                       // xl (GEMM output) / head z
    float* big1   = ws + BIG;                 // attention-aggregated out
    float* big2   = ws + 2 * BIG;             // BN+ELU output = next layer input
    float* sbuf   = ws + 3 * BIG;
    float* dbuf   = sbuf   + (long)NN * 4;
    float* emax   = dbuf   + (long)NN * 4;
    float* denom  = emax   + (long)NN * 4;
    float* xpad   = denom  + (long)NN * 4;    // [20000,32]
    float* w0pad  = xpad   + (long)NN * 32;   // [32,512]
    float* hw2pad = w0pad  + 32L * 512;       // [128,16]
    float* hb2pad = hw2pad + 128L * 16;       // [16]
    float* outtmp = hb2pad + 16;              // [20000,16]

    // ---- pad awkward shapes so every GEMM is branch-free ----
    pad2d<<<(int)(((long)NN * 32 + 255) / 256), 256, 0, stream>>>(x, xpad, NN, 29, NN, 32);
    pad2d<<<(32 * 512 + 255) / 256, 256, 0, stream>>>(W[0], w0pad, 29, 512, 32, 512);
    pad2d<<<(128 * 16 + 255) / 256, 256, 0, stream>>>(hw2, hw2pad, 128, 9, 128, 16);
    pad2d<<<1, 64, 0, stream>>>(hb2, hb2pad, 1, 9, 1, 16);

    const dim3 gN512(512 / 64, NN / 32);      // 32x64 tile per wave
    const dim3 gN128(128 / 64, NN / 32);
    const dim3 gN16 (1,        NN / 32);

    // ---- layer 0 : K=32 (padded), H=4 ----
    gemm_wmma<32, 512, 2, 4><<<gN512, 32, 0, stream>>>(xpad, w0pad, big0, nullptr, 0);
    run_attention(ei, asr[0], ads[0], bia[0], gg[0], bb[0], mm[0], vv[0],
                  big0, big1, big2, sbuf, dbuf, emax, denom, 4, stream);

    // ---- layer 1 : K=512, H=4 ----
    gemm_wmma<512, 512, 2, 4><<<gN512, 32, 0, stream>>>(big2, W[1], big0, nullptr, 0);
    run_attention(ei, asr[1], ads[1], bia[1], gg[1], bb[1], mm[1], vv[1],
                  big0, big1, big2, sbuf, dbuf, emax, denom, 4, stream);

    // ---- layer 2 : K=512, H=1 ----
    gemm_wmma<512, 128, 2, 4><<<gN128, 32, 0, stream>>>(big2, W[2], big0, nullptr, 0);
    run_attention(ei, asr[2], ads[2], bia[2], gg[2], bb[2], mm[2], vv[2],
                  big0, big1, big2, sbuf, dbuf, emax, denom, 1, stream);

    // ---- head: z = relu(h3 @ hw1 + hb1) -> big0 ; logits -> outtmp ----
    gemm_wmma<128, 128, 2, 4><<<gN128, 32, 0, stream>>>(big2, hw1, big0, hb1, 1);
    gemm_wmma<128, 16, 2, 1><<<gN16, 32, 0, stream>>>(big0, hw2pad, outtmp, hb2pad, 0);
    slice9<<<(int)(((long)NN * 9 + 255) / 256), 256, 0, stream>>>(outtmp, (float*)d_out);
}